// BiDecoder_7739531067738
// MI455X (gfx1250) — compile-verified
//
#include <hip/hip_runtime.h>

// Per-edge gather + dot:  out[e] = dot(ufeat[src[e]], ifeat[dst[e]]), D = 256.
// Memory-bound (AI ~0.25 FLOP/B); gather working set ~72MB -> L2-resident on MI455X.
// Each wave batches 16 edges through V_WMMA_F32_16X16X4_F32 (C[i][j] = <hu_i, hv_j>,
// diagonal = answers). Redundant off-diagonal FLOPs are free at this roofline.

typedef float v2f __attribute__((ext_vector_type(2)));
typedef float v8f __attribute__((ext_vector_type(8)));

#define FEAT_D 256
#define EDGES_PER_WAVE 16
#define WAVES_PER_BLOCK 8
#define EDGES_PER_BLOCK (EDGES_PER_WAVE * WAVES_PER_BLOCK)

__global__ __launch_bounds__(256) void edge_dot_wmma_f32(
    const float* __restrict__ ufeat,   // (N_GENE, 256)
    const float* __restrict__ ifeat,   // (N_CELL, 256)
    const int*   __restrict__ src,     // (E,)
    const int*   __restrict__ dst,     // (E,)
    float*       __restrict__ out,     // (E, 1)
    int E)
{
    const int lane  = threadIdx.x & 31;
    const int wave  = threadIdx.x >> 5;
    const int ebase = (blockIdx.x * WAVES_PER_BLOCK + wave) * EDGES_PER_WAVE;

    // Wave-uniform early out keeps EXEC all-1s for every executed WMMA.
    if (ebase >= E) return;

    const int m    = lane & 15;   // edge slot: A row index / B column index
    const int half = lane >> 4;   // 0 -> K+{0,1}, 1 -> K+{2,3} per 16x4 f32 layout

    // Clamp tail edges so all 32 lanes issue valid loads (EXEC must stay full).
    int e  = ebase + m;
    int ec = (e < E) ? e : (E - 1);

    // Row bases (max offset 50000*256 < 2^31, int math is safe).
    const float* up = ufeat + src[ec] * FEAT_D + 2 * half;
    const float* vp = ifeat + dst[ec] * FEAT_D + 2 * half;

    v8f c = {};
#pragma unroll 8
    for (int k0 = 0; k0 < FEAT_D; k0 += 4) {
        // A 16x4: lane m half h supplies {hu_m[k0+2h], hu_m[k0+2h+1]}
        v2f a = *(const v2f*)(up + k0);
        // B 4x16: lane n half h supplies {hv_n[k0+2h], hv_n[k0+2h+1]}
        v2f b = *(const v2f*)(vp + k0);
        // 8 args: (neg_a, A, neg_b, B, c_mod, C, reuse_a, reuse_b)
        c = __builtin_amdgcn_wmma_f32_16x16x4_f32(
                false, a, false, b, (short)0, c, false, false);
    }

    // Diagonal extraction (16x16 f32 C/D layout):
    //   d in 0..7  -> lane d      , VGPR d
    //   d in 8..15 -> lane d + 16 , VGPR d - 8
    // => writing lanes are 0..7 (half 0) and 24..31 (half 1).
    const int idx = lane & 15;            // output edge slot d
    if (half == (idx >> 3)) {
        const int r = idx & 7;            // accumulator VGPR holding C[d][d]
        float val = c[0];
        if (r == 1) val = c[1];
        if (r == 2) val = c[2];
        if (r == 3) val = c[3];
        if (r == 4) val = c[4];
        if (r == 5) val = c[5];
        if (r == 6) val = c[6];
        if (r == 7) val = c[7];
        const int eo = ebase + idx;
        if (eo < E) out[eo] = val;
    }
}

extern "C" void kernel_launch(void* const* d_in, const int* in_sizes, int n_in,
                              void* d_out, int out_size, void* d_ws, size_t ws_size,
                              hipStream_t stream) {
    const float* ufeat = (const float*)d_in[0];
    const float* ifeat = (const float*)d_in[1];
    const int*   src   = (const int*)d_in[2];
    const int*   dst   = (const int*)d_in[3];
    float*       out   = (float*)d_out;

    const int E = in_sizes[2];   // number of edges
    const int blocks = (E + EDGES_PER_BLOCK - 1) / EDGES_PER_BLOCK;

    edge_dot_wmma_f32<<<dim3(blocks), dim3(256), 0, stream>>>(
        ufeat, ifeat, src, dst, out, E);
}